// SingleHGCN_47081431499245
// MI455X (gfx1250) — compile-verified
//
#include <hip/hip_runtime.h>
#include <hip/hip_bf16.h>
#include <math.h>

// ---------------------------------------------------------------------------
// Problem constants (match reference)
// ---------------------------------------------------------------------------
#define NN    8192     // nodes / hyperedges
#define DF    784      // feature dim
#define DE    256      // embedding dim
#define TK    11       // top_k + 1

typedef __attribute__((ext_vector_type(2))) float v2f;
typedef __attribute__((ext_vector_type(8))) float v8f;
typedef __attribute__((ext_vector_type(4))) int   v4i;

// gfx1250 async global->LDS path (ASYNCcnt), with sync fallback
#if defined(__has_builtin)
#if __has_builtin(__builtin_amdgcn_global_load_async_to_lds_b128) && \
    __has_builtin(__builtin_amdgcn_s_wait_asynccnt)
#define USE_ASYNC_LDS 1
#endif
#endif
#ifndef USE_ASYNC_LDS
#define USE_ASYNC_LDS 0
#endif

typedef __attribute__((address_space(1))) v4i as1_v4i;   // global int4
typedef __attribute__((address_space(3))) v4i as3_v4i;   // LDS int4

__device__ __forceinline__ void stage_b128(const float* __restrict__ g,
                                           float* __restrict__ l) {
#if USE_ASYNC_LDS
    __builtin_amdgcn_global_load_async_to_lds_b128((as1_v4i*)g, (as3_v4i*)l, 0, 0);
#else
    const float4 v = *(const float4*)g;
    l[0] = v.x; l[1] = v.y; l[2] = v.z; l[3] = v.w;
#endif
}

__device__ __forceinline__ void wait_async() {
#if USE_ASYNC_LDS
    __builtin_amdgcn_s_wait_asynccnt(0);
#endif
}

// ---------------------------------------------------------------------------
// Kernel 1: row squared norms  sq[i] = sum_c X[i,c]^2
// ---------------------------------------------------------------------------
__global__ __launch_bounds__(256)
void row_sq_kernel(const float* __restrict__ X, float* __restrict__ sq, int D) {
    const int row = blockIdx.x;
    float acc = 0.f;
    for (int c = threadIdx.x; c < D; c += 256) {
        float v = X[(size_t)row * D + c];
        acc += v * v;
    }
    __shared__ float red[256];
    red[threadIdx.x] = acc;
    __syncthreads();
    for (int s = 128; s > 0; s >>= 1) {
        if (threadIdx.x < s) red[threadIdx.x] += red[threadIdx.x + s];
        __syncthreads();
    }
    if (threadIdx.x == 0) sq[row] = red[0];
}

// ---------------------------------------------------------------------------
// Tiled WMMA fp32 GEMM (V_WMMA_F32_16X16X4_F32), 64x64 block tile, 8 waves,
// double-buffered LDS with async global->LDS staging (one barrier / K-tile).
// MODE 1: C[m,n] = |sq[m] + sq[n] - 2*(A·B^T)[m,n]|   (B row-major [N,K])
// MODE 0: C[m,n] = rowScale * (A·B)[m,n]              (B row-major [K,N])
// Requires M%64==0, N%64==0, K%16==0 (true for all uses here).
// ---------------------------------------------------------------------------
#define LDSTR 20   // 80B row stride: 16B-aligned B128 slots AND 20*r%64 distinct

template <int MODE>
__global__ __launch_bounds__(256)
void wmma_gemm_kernel(const float* __restrict__ A, const float* __restrict__ B,
                      float* __restrict__ C, int M, int N, int K,
                      int lda, int ldb, int ldc,
                      const float* __restrict__ sq, float rowScale) {
    __shared__ float As[2][64][LDSTR];   // [buf][m][k]
    __shared__ float Bs[2][64][LDSTR];   // [buf][n][k]

    const int tid   = threadIdx.x;
    const int wave  = tid >> 5;          // 0..7
    const int lane  = tid & 31;
    const int mt    = wave & 3;          // M subtile 0..3
    const int npair = wave >> 2;         // 0..1 -> N subtiles {2*npair, 2*npair+1}
    const int mbase = blockIdx.y * 64;
    const int nbase = blockIdx.x * 64;

    const int lm = lane & 15;            // 16-dim index (M for A, N for B)
    const int kb = (lane >> 4) << 1;     // lanes 16-31 hold K=2,3 of each group

    v8f acc[2];
    acc[0] = (v8f)0.f;
    acc[1] = (v8f)0.f;

    // staging indices
    const int lr = tid >> 2;             // 0..63
    const int lk = (tid & 3) << 2;       // 0,4,8,12
    const int bk = tid >> 4;             // 0..15  (MODE 0 B stage)
    const int bn = (tid & 15) << 2;      // 0..60

    const int nk = K / 16;

    auto stage = [&](int t, int buf) {
        const int k0 = t * 16;
        // A tile: 64 rows x 16 K, one B128 per thread
        stage_b128(&A[(size_t)(mbase + lr) * lda + k0 + lk], &As[buf][lr][lk]);
        if (MODE == 1) {
            // B tile from row-major [N,K]: contiguous B128 per thread
            stage_b128(&B[(size_t)(nbase + lr) * ldb + k0 + lk], &Bs[buf][lr][lk]);
        } else {
            // B tile from row-major [K,N]: transposed scatter (sync path)
            const float4 bv = *(const float4*)&B[(size_t)(k0 + bk) * ldb + nbase + bn];
            Bs[buf][bn + 0][bk] = bv.x; Bs[buf][bn + 1][bk] = bv.y;
            Bs[buf][bn + 2][bk] = bv.z; Bs[buf][bn + 3][bk] = bv.w;
        }
    };

    stage(0, 0);

    for (int t = 0; t < nk; ++t) {
        const int buf = t & 1;
        wait_async();          // my async writes into `buf` have landed
        __syncthreads();       // everyone's writes visible; prev buf fully consumed
        if (t + 1 < nk) stage(t + 1, buf ^ 1);   // overlap with compute below

        #pragma unroll
        for (int kk = 0; kk < 16; kk += 4) {
            v2f a;
            a.x = As[buf][mt * 16 + lm][kk + kb];
            a.y = As[buf][mt * 16 + lm][kk + kb + 1];
            #pragma unroll
            for (int s = 0; s < 2; ++s) {
                const int nt = npair * 2 + s;
                v2f b;
                b.x = Bs[buf][nt * 16 + lm][kk + kb];
                b.y = Bs[buf][nt * 16 + lm][kk + kb + 1];
                acc[s] = __builtin_amdgcn_wmma_f32_16x16x4_f32(
                    false, a, false, b, (short)0, acc[s], false, false);
            }
        }
    }

    // ---- epilogue: C/D layout  m = r + 8*(lane>>4), n = lane&15 ----
    const int gmBase = mbase + mt * 16 + ((lane >> 4) << 3);
    #pragma unroll
    for (int s = 0; s < 2; ++s) {
        const int nt = npair * 2 + s;
        const int gn = nbase + nt * 16 + lm;
        #pragma unroll
        for (int r = 0; r < 8; ++r) {
            const int gm = gmBase + r;
            const float v = acc[s][r];
            if (MODE == 1) {
                C[(size_t)gm * ldc + gn] = fabsf(sq[gm] + sq[gn] - 2.0f * v);
            } else {
                C[(size_t)gm * ldc + gn] = rowScale * v;
            }
        }
    }
}

// ---------------------------------------------------------------------------
// Kernel: top-K smallest per row (tie -> smaller index), 1 block = 1 row.
// Full row staged in LDS (32KB of the 320KB/WGP), K argmin+mask passes.
// ---------------------------------------------------------------------------
__global__ __launch_bounds__(256)
void topk_kernel(const float* __restrict__ dist, int* __restrict__ idx) {
    const int row = blockIdx.x;
    const int tid = threadIdx.x;
    __shared__ float vals[NN];
    __shared__ float rv[256];
    __shared__ int   ri[256];

    for (int c = tid; c < NN; c += 256)
        vals[c] = dist[(size_t)row * NN + c];
    __syncthreads();

    for (int k = 0; k < TK; ++k) {
        float best = __builtin_huge_valf();
        int   bi   = NN;
        for (int c = tid; c < NN; c += 256) {
            float v = vals[c];
            if (v < best) { best = v; bi = c; }   // c strictly increasing per thread
        }
        rv[tid] = best; ri[tid] = bi;
        __syncthreads();
        for (int s = 128; s > 0; s >>= 1) {
            if (tid < s) {
                float v2 = rv[tid + s]; int i2 = ri[tid + s];
                if (v2 < rv[tid] || (v2 == rv[tid] && i2 < ri[tid])) {
                    rv[tid] = v2; ri[tid] = i2;
                }
            }
            __syncthreads();
        }
        if (tid == 0) {
            idx[row * TK + k] = ri[0];
            vals[ri[0]] = __builtin_huge_valf();
        }
        __syncthreads();
    }
}

// ---------------------------------------------------------------------------
// Utility kernels
// ---------------------------------------------------------------------------
__global__ void zero_f32_kernel(float* __restrict__ p, size_t n) {
    size_t i = (size_t)blockIdx.x * blockDim.x + threadIdx.x;
    const size_t stride = (size_t)gridDim.x * blockDim.x;
    for (; i < n; i += stride) p[i] = 0.0f;
}

__global__ void scatter_h_kernel(const int* __restrict__ idx, float* __restrict__ H) {
    const int t = blockIdx.x * blockDim.x + threadIdx.x;
    if (t >= NN * TK) return;
    const int i = t / TK;
    const int r = idx[t];
    H[(size_t)r * NN + i] = 1.0f;
}

__global__ void count_rows_kernel(const int* __restrict__ idx, int* __restrict__ cnt) {
    const int t = blockIdx.x * blockDim.x + threadIdx.x;
    if (t < NN * TK) atomicAdd(&cnt[idx[t]], 1);
}

__global__ void dv_kernel(const int* __restrict__ cnt, float* __restrict__ dv) {
    const int t = blockIdx.x * blockDim.x + threadIdx.x;
    if (t < NN) dv[t] = rsqrtf((float)cnt[t]);   // cnt>=1 (self always in own top-k)
}

// M1[i,:] = sum_k dv[j]*X[j,:], j = idx[i,k]     ( = H^T @ (Dv X) row i )
__global__ __launch_bounds__(256)
void gather_scale_kernel(const float* __restrict__ X, const int* __restrict__ idx,
                         const float* __restrict__ dv, float* __restrict__ M1) {
    const int i = blockIdx.x;
    __shared__ int   js[TK];
    __shared__ float ds[TK];
    if (threadIdx.x < TK) {
        const int j = idx[i * TK + threadIdx.x];
        js[threadIdx.x] = j;
        ds[threadIdx.x] = dv[j];
    }
    __syncthreads();
    for (int c = threadIdx.x; c < DF; c += 256) {
        float acc = 0.f;
        #pragma unroll
        for (int k = 0; k < TK; ++k)
            acc += ds[k] * X[(size_t)js[k] * DF + c];
        M1[(size_t)i * DF + c] = acc;
    }
}

// E_acc[idx[i,k], :] += deC * Xout[i, :]  (block = one (i,k) pair, 256 cols)
__global__ __launch_bounds__(256)
void scatter_e_kernel(const float* __restrict__ Xout, const int* __restrict__ idx,
                      float* __restrict__ E, float deC) {
    const int t = blockIdx.x;        // 0 .. NN*TK-1
    const int i = t / TK;
    const int r = idx[t];
    const int c = threadIdx.x;
    atomicAdd(&E[(size_t)r * DE + c], deC * Xout[(size_t)i * DE + c]);
}

__global__ __launch_bounds__(256)
void scale_e_kernel(float* __restrict__ E, const float* __restrict__ dv) {
    const int r = blockIdx.x;
    const int c = threadIdx.x;
    E[(size_t)r * DE + c] *= dv[r];
}

// ---------------------------------------------------------------------------
// Launch
// ---------------------------------------------------------------------------
extern "C" void kernel_launch(void* const* d_in, const int* in_sizes, int n_in,
                              void* d_out, int out_size, void* d_ws, size_t ws_size,
                              hipStream_t stream) {
    const float* X     = (const float*)d_in[0];   // [8192, 784]
    const float* theta = (const float*)d_in[1];   // [784, 256]

    float* Xout = (float*)d_out;                              // [8192, 256]
    float* E    = Xout + (size_t)NN * DE;                     // [8192, 256]
    float* H    = E + (size_t)NN * DE;                        // [8192, 8192] (also dist scratch)

    // workspace carve-up (256B aligned)
    char* ws = (char*)d_ws;
    float* sq     = (float*)(ws + 0);                         // 32 KB
    float* dv     = (float*)(ws + (32 << 10));                // 32 KB
    int*   rowcnt = (int*)  (ws + (64 << 10));                // 32 KB
    int*   idx    = (int*)  (ws + (96 << 10));                // 8192*11*4 = 352.4 KB
    float* M1     = (float*)(ws + (96 << 10) + 360448);       // 8192*784*4 = 25.7 MB

    const float deC = 1.0f / sqrtf((float)TK);                // column degree is exactly TK

    // 1) row norms
    row_sq_kernel<<<NN, 256, 0, stream>>>(X, sq, DF);

    // 2) dist = |sq_i + sq_j - 2 X X^T|  -> written into H region (scratch)
    wmma_gemm_kernel<1><<<dim3(NN / 64, NN / 64), 256, 0, stream>>>(
        X, X, H, NN, NN, DF, DF, DF, NN, sq, 0.f);

    // 3) top-11 smallest per row (includes self at ~0)
    topk_kernel<<<NN, 256, 0, stream>>>(H, idx);

    // 4) rebuild H as 0/1 incidence
    zero_f32_kernel<<<4096, 256, 0, stream>>>(H, (size_t)NN * NN);
    scatter_h_kernel<<<(NN * TK + 255) / 256, 256, 0, stream>>>(idx, H);

    // 5) vertex degrees -> dv = deg^-1/2
    zero_f32_kernel<<<32, 256, 0, stream>>>((float*)rowcnt, NN);
    count_rows_kernel<<<(NN * TK + 255) / 256, 256, 0, stream>>>(idx, rowcnt);
    dv_kernel<<<NN / 256, 256, 0, stream>>>(rowcnt, dv);

    // 6) M1 = H^T @ (Dv X)   (11-row gather per output row)
    gather_scale_kernel<<<NN, 256, 0, stream>>>(X, idx, dv, M1);

    // 7) X_out = deC * (M1 @ theta)
    wmma_gemm_kernel<0><<<dim3(DE / 64, NN / 64), 256, 0, stream>>>(
        M1, theta, Xout, NN, DE, DF, DF, DE, DE, nullptr, deC);

    // 8) E = Dv @ (H @ (deC * X_out)) via scatter-add then row scale
    zero_f32_kernel<<<2048, 256, 0, stream>>>(E, (size_t)NN * DE);
    scatter_e_kernel<<<NN * TK, 256, 0, stream>>>(Xout, idx, E, deC);
    scale_e_kernel<<<NN, 256, 0, stream>>>(E, dv);
}